// JODIE_13838384628052
// MI455X (gfx1250) — compile-verified
//
#include <hip/hip_runtime.h>
#include <hip/hip_bf16.h>

typedef __attribute__((ext_vector_type(2))) float v2f;
typedef __attribute__((ext_vector_type(4))) float v4f;
typedef __attribute__((ext_vector_type(8))) float v8f;

#define NNODES 200000
#define S4     4000
#define R3     12000
#define DEMB   128
#define DNODE  172
#define DEDGE  172
#define DTIME  100
#define DMAIL  300
#define DIN    400
#define KXH    528           // 400 (x) + 128 (h)
#define XH_STRIDE 529        // odd mod-64 stride -> conflict-free LDS
#define NF_STRIDE 173
#define A_STRIDE  132        // 16B-aligned rows for async B128; banks 4m+k conflict-free

// ---------------------------------------------------------------- copies
__global__ void k_copy(const v4f* __restrict__ mem, const v4f* __restrict__ mail,
                       v4f* __restrict__ out_mem, v4f* __restrict__ out_mail) {
  long stride = (long)gridDim.x * blockDim.x;
  long idx = (long)blockIdx.x * blockDim.x + threadIdx.x;
  const long nm = (long)NNODES * DEMB / 4;
  const long nl = (long)NNODES * DMAIL / 4;
  // mem table: keep default (RT) policy -- 102 MB fits the 192 MB L2 and is
  // gathered again by k_embed / k_mail.
  for (long i = idx; i < nm; i += stride) out_mem[i] = mem[i];
  // mail table: 240 MB each way, never re-used -> stream non-temporally.
  for (long i = idx; i < nl; i += stride)
    __builtin_nontemporal_store(__builtin_nontemporal_load(&mail[i]), &out_mail[i]);
}

// ------------------------------------------------- deterministic scatter prep
__global__ void k_init_win(int* w1, int* w2) {
  int i = blockIdx.x * blockDim.x + threadIdx.x;
  if (i < NNODES) { w1[i] = -1; w2[i] = -1; }
}

__global__ void k_winner(const int* __restrict__ nodes, const int* __restrict__ adj_dst,
                         int* w1, int* w2) {
  int i = blockIdx.x * blockDim.x + threadIdx.x;
  if (i < 2 * S4) {
    atomicMax(&w1[nodes[i]], i);
    atomicMax(&w2[adj_dst[i]], i);
  }
}

// ------------------------------------------------- weight transposes (k-major B)
__global__ void k_transpose(const float* __restrict__ Wih, const float* __restrict__ Whh,
                            const float* __restrict__ nodeW,
                            const float* __restrict__ Wsrc, const float* __restrict__ Wdst,
                            float* Bt1, float* Bt2, float* WsT, float* WdT) {
  const int n1 = KXH * DEMB;
  const int n2 = DNODE * DEMB;
  const int n3 = DEMB * DEMB;
  const int tot = n1 + n2 + 2 * n3;
  for (int i = blockIdx.x * blockDim.x + threadIdx.x; i < tot; i += gridDim.x * blockDim.x) {
    if (i < n1) {
      int k = i / DEMB, o = i % DEMB;
      Bt1[i] = (k < DIN) ? Wih[o * DIN + k] : Whh[o * DEMB + (k - DIN)];
    } else if (i < n1 + n2) {
      int j = i - n1, k = j / DEMB, o = j % DEMB;
      Bt2[j] = nodeW[o * DNODE + k];
    } else if (i < n1 + n2 + n3) {
      int j = i - n1 - n2, k = j / DEMB, o = j % DEMB;
      WsT[j] = Wsrc[o * DEMB + k];
    } else {
      int j = i - n1 - n2 - n3, k = j / DEMB, o = j % DEMB;
      WdT[j] = Wdst[o * DEMB + k];
    }
  }
}

// ------------------------------------------------- fused RNN + node-fc + LN + projection
__global__ __launch_bounds__(256) void k_embed(
    const float* __restrict__ mem_data, const float* __restrict__ mem_time,
    const float* __restrict__ mail, const float* __restrict__ mail_time,
    const float* __restrict__ nfeat, const float* __restrict__ times,
    const float* __restrict__ te_w, const float* __restrict__ bih,
    const float* __restrict__ bhh, const float* __restrict__ node_b,
    const float* __restrict__ ln_g, const float* __restrict__ ln_b,
    const float* __restrict__ tl_W, const float* __restrict__ tl_b,
    const int* __restrict__ nodes, const float* __restrict__ Bt1,
    const float* __restrict__ Bt2, const int* __restrict__ win1,
    float* __restrict__ out_mem, float* __restrict__ e_proj) {
  __shared__ float xh[16 * XH_STRIDE];
  __shared__ float nf[16 * NF_STRIDE];
  __shared__ float psum[8][16];
  __shared__ float psq[8][16];
  __shared__ int   s_node[16];
  __shared__ float s_dt[16];
  __shared__ float s_td[16];

  const int tid = threadIdx.x;
  const int wg = blockIdx.x;

  if (tid < 16) {
    int i = wg * 16 + tid;
    int n = nodes[i];
    float mt = mail_time[n];
    s_node[tid] = n;
    s_dt[tid] = mt - mem_time[n];
    float t = times[i];
    s_td[tid] = (t - mt) / (t + 1.0f);
  }
  __syncthreads();

  // stage x||h (16x528) and nfeat (16x172) tiles; 16 consecutive lanes per row -> coalesced
  {
    int r = tid >> 4;
    int c0 = tid & 15;
    int n = s_node[r];
    float dt = s_dt[r];
    for (int c = c0; c < KXH; c += 16) {
      float v;
      if (c < DMAIL)       v = mail[(long)n * DMAIL + c];
      else if (c < DIN)    v = __cosf(dt * te_w[c - DMAIL]);
      else                 v = mem_data[(long)n * DEMB + (c - DIN)];
      xh[r * XH_STRIDE + c] = v;
    }
    for (int c = c0; c < DNODE; c += 16)
      nf[r * NF_STRIDE + c] = nfeat[(long)n * DNODE + c];
  }
  __syncthreads();

  const int lane = tid & 31;
  const int w = tid >> 5;          // wave id: owns output cols [16w,16w+16)
  const int m = lane & 15;
  const int kh = lane >> 4;
  const int col = w * 16 + m;

  // FP32 WMMA, K-step 4:  acc1 = (x||h) @ [Wih|Whh]^T tile
  v8f acc1 = {};
  for (int k0 = 0; k0 < KXH; k0 += 4) {
    int ka = k0 + 2 * kh;
    v2f a, b;
    a.x = xh[m * XH_STRIDE + ka];
    a.y = xh[m * XH_STRIDE + ka + 1];
    b.x = Bt1[ka * DEMB + col];
    b.y = Bt1[(ka + 1) * DEMB + col];
    acc1 = __builtin_amdgcn_wmma_f32_16x16x4_f32(false, a, false, b, (short)0, acc1,
                                                 false, false);
  }
  // acc2 = nfeat @ node_W^T tile
  v8f acc2 = {};
  for (int k0 = 0; k0 < DNODE; k0 += 4) {
    int ka = k0 + 2 * kh;
    v2f a, b;
    a.x = nf[m * NF_STRIDE + ka];
    a.y = nf[m * NF_STRIDE + ka + 1];
    b.x = Bt2[ka * DEMB + col];
    b.y = Bt2[(ka + 1) * DEMB + col];
    acc2 = __builtin_amdgcn_wmma_f32_16x16x4_f32(false, a, false, b, (short)0, acc2,
                                                 false, false);
  }

  const float bias1 = bih[col] + bhh[col];
  const float bias2 = node_b[col];
  float ev[8];
#pragma unroll
  for (int v = 0; v < 8; v++)
    ev[v] = tanhf(acc1[v] + bias1) + acc2[v] + bias2;

  // deterministic LayerNorm reduce: shuffle within 16-lane half, then 8 wave partials in LDS
#pragma unroll
  for (int v = 0; v < 8; v++) {
    float s = ev[v], s2 = ev[v] * ev[v];
    for (int off = 1; off < 16; off <<= 1) {
      s  += __shfl_xor(s, off, 32);
      s2 += __shfl_xor(s2, off, 32);
    }
    if (m == 0) {
      psum[w][v + 8 * kh] = s;
      psq[w][v + 8 * kh] = s2;
    }
  }
  __syncthreads();

  const float gln = ln_g[col], bln = ln_b[col];
  const float tw = tl_W[col], tb = tl_b[col];
#pragma unroll
  for (int v = 0; v < 8; v++) {
    int row = v + 8 * kh;
    float su = 0.f, sq = 0.f;
#pragma unroll
    for (int ww = 0; ww < 8; ww++) { su += psum[ww][row]; sq += psq[ww][row]; }
    float mu = su * (1.0f / 128.0f);
    float var = sq * (1.0f / 128.0f) - mu * mu;
    float rs = rsqrtf(var + 1e-5f);
    float nrm = (ev[v] - mu) * rs * gln + bln;
    int i = wg * 16 + row;
    if (i < 2 * S4) {                 // memory scatter (last-index-wins)
      int nn = s_node[row];
      if (win1[nn] == i) out_mem[(long)nn * DEMB + col] = nrm;
    }
    float proj = nrm * (1.0f + s_td[row] * tw + tb);
    e_proj[(long)i * DEMB + col] = proj;
  }
}

// ------------------------------------------------- edge predictor
__global__ __launch_bounds__(256) void k_pred(
    const float* __restrict__ e_proj, const float* __restrict__ WsT,
    const float* __restrict__ WdT, const float* __restrict__ bsrc,
    const float* __restrict__ bdst, const float* __restrict__ wout,
    const float* __restrict__ bout, float* __restrict__ scores) {
  __shared__ float As[16 * A_STRIDE];
  __shared__ float Ad[16 * A_STRIDE];
  __shared__ float part[8][16];

  const int tid = threadIdx.x;
  const int half = blockIdx.x / 250;     // 0=pos, 1=neg
  const int tile = blockIdx.x % 250;

  // Async-DMA the two 16x128 e_proj tiles into LDS (B128 chunks, ASYNCcnt-tracked,
  // no VGPR staging). 512 chunks per tile, 2 per thread per tile.
  {
    int j = tid;
#pragma unroll
    for (int q = 0; q < 2; ++q, j += 256) {
      int row = j >> 5;                 // 32 B128 chunks per 128-float row
      int c4 = (j & 31) << 2;           // float column, multiple of 4
      int src_i = tile * 16 + row;
      int dst_i = src_i + (half ? 2 * S4 : S4);
      unsigned lds_s = (unsigned)(size_t)&As[row * A_STRIDE + c4];
      unsigned lds_d = (unsigned)(size_t)&Ad[row * A_STRIDE + c4];
      unsigned off_s = (unsigned)(((long)src_i * DEMB + c4) * 4);
      unsigned off_d = (unsigned)(((long)dst_i * DEMB + c4) * 4);
      asm volatile("global_load_async_to_lds_b128 %0, %1, %2"
                   :: "v"(lds_s), "v"(off_s), "s"(e_proj) : "memory");
      asm volatile("global_load_async_to_lds_b128 %0, %1, %2"
                   :: "v"(lds_d), "v"(off_d), "s"(e_proj) : "memory");
    }
    asm volatile("s_wait_asynccnt 0" ::: "memory");
  }
  __syncthreads();

  const int lane = tid & 31;
  const int w = tid >> 5;
  const int m = lane & 15;
  const int kh = lane >> 4;
  const int col = w * 16 + m;

  v8f accs = {}, accd = {};
  for (int k0 = 0; k0 < DEMB; k0 += 4) {
    int ka = k0 + 2 * kh;
    v2f a, b;
    a.x = As[m * A_STRIDE + ka];
    a.y = As[m * A_STRIDE + ka + 1];
    b.x = WsT[ka * DEMB + col];
    b.y = WsT[(ka + 1) * DEMB + col];
    accs = __builtin_amdgcn_wmma_f32_16x16x4_f32(false, a, false, b, (short)0, accs,
                                                 false, false);
    a.x = Ad[m * A_STRIDE + ka];
    a.y = Ad[m * A_STRIDE + ka + 1];
    b.x = WdT[ka * DEMB + col];
    b.y = WdT[(ka + 1) * DEMB + col];
    accd = __builtin_amdgcn_wmma_f32_16x16x4_f32(false, a, false, b, (short)0, accd,
                                                 false, false);
  }

  const float bb = bsrc[col] + bdst[col];
  const float wo = wout[col];
#pragma unroll
  for (int v = 0; v < 8; v++) {
    float h = accs[v] + accd[v] + bb;
    h = fmaxf(h, 0.0f);
    float s = h * wo;
    for (int off = 1; off < 16; off <<= 1) s += __shfl_xor(s, off, 32);
    if (m == 0) part[w][v + 8 * kh] = s;
  }
  __syncthreads();
  if (tid < 16) {
    float s = 0.f;
#pragma unroll
    for (int ww = 0; ww < 8; ww++) s += part[ww][tid];
    scores[half * S4 + tile * 16 + tid] = s + bout[0];
  }
}

// ------------------------------------------------- mailbox scatter
__global__ void k_mail(const int* __restrict__ adj_src, const int* __restrict__ adj_dst,
                       const int* __restrict__ eids, const float* __restrict__ efeat,
                       const int* __restrict__ win2, const float* __restrict__ new_mem,
                       float* __restrict__ new_mail) {
  int i = blockIdx.x;
  int d = adj_dst[i];
  if (win2[d] != i) return;
  int s = adj_src[i];
  int e = eids[i];
  for (int c = threadIdx.x; c < DMAIL; c += blockDim.x) {
    float v = (c < DEMB) ? new_mem[(long)s * DEMB + c]
                         : efeat[(long)e * DEDGE + (c - DEMB)];
    __builtin_nontemporal_store(v, &new_mail[(long)d * DMAIL + c]);
  }
}

// ================================================================ launch
extern "C" void kernel_launch(void* const* d_in, const int* in_sizes, int n_in,
                              void* d_out, int out_size, void* d_ws, size_t ws_size,
                              hipStream_t stream) {
  (void)in_sizes; (void)n_in; (void)out_size; (void)ws_size;
  const float* mem_data  = (const float*)d_in[0];
  const float* mem_time  = (const float*)d_in[1];
  const float* mail      = (const float*)d_in[2];
  const float* mail_time = (const float*)d_in[3];
  const float* nfeat     = (const float*)d_in[4];
  const float* efeat     = (const float*)d_in[5];
  const float* times     = (const float*)d_in[6];
  const float* te_w      = (const float*)d_in[7];
  const float* rnn_Wih   = (const float*)d_in[8];
  const float* rnn_Whh   = (const float*)d_in[9];
  const float* rnn_bih   = (const float*)d_in[10];
  const float* rnn_bhh   = (const float*)d_in[11];
  const float* node_W    = (const float*)d_in[12];
  const float* node_b    = (const float*)d_in[13];
  const float* ln_g      = (const float*)d_in[14];
  const float* ln_b      = (const float*)d_in[15];
  const float* tl_W      = (const float*)d_in[16];
  const float* tl_b      = (const float*)d_in[17];
  const float* ep_Wsrc   = (const float*)d_in[18];
  const float* ep_bsrc   = (const float*)d_in[19];
  const float* ep_Wdst   = (const float*)d_in[20];
  const float* ep_bdst   = (const float*)d_in[21];
  const float* ep_Wout   = (const float*)d_in[22];
  const float* ep_bout   = (const float*)d_in[23];
  const int*   nodes     = (const int*)d_in[24];
  const int*   adj_src   = (const int*)d_in[25];
  const int*   adj_dst   = (const int*)d_in[26];
  const int*   eids      = (const int*)d_in[27];

  float* out = (float*)d_out;
  float* scores   = out;                                   // [8000]
  float* new_mem  = out + 8000;                            // [N*128]
  float* new_mail = new_mem + (long)NNODES * DEMB;         // [N*300]

  float* ws = (float*)d_ws;
  float* Bt1 = ws;                                         // [528*128]
  float* Bt2 = Bt1 + KXH * DEMB;                           // [172*128]
  float* WsT = Bt2 + DNODE * DEMB;                         // [128*128]
  float* WdT = WsT + DEMB * DEMB;                          // [128*128]
  int*   win1 = (int*)(WdT + DEMB * DEMB);                 // [N]
  int*   win2 = win1 + NNODES;                             // [N]
  float* e_proj = (float*)(win2 + NNODES);                 // [12000*128]

  k_copy<<<2048, 256, 0, stream>>>((const v4f*)mem_data, (const v4f*)mail,
                                   (v4f*)new_mem, (v4f*)new_mail);
  k_init_win<<<(NNODES + 255) / 256, 256, 0, stream>>>(win1, win2);
  k_winner<<<(2 * S4 + 255) / 256, 256, 0, stream>>>(nodes, adj_dst, win1, win2);
  k_transpose<<<512, 256, 0, stream>>>(rnn_Wih, rnn_Whh, node_W, ep_Wsrc, ep_Wdst,
                                       Bt1, Bt2, WsT, WdT);
  k_embed<<<R3 / 16, 256, 0, stream>>>(mem_data, mem_time, mail, mail_time, nfeat,
                                       times, te_w, rnn_bih, rnn_bhh, node_b, ln_g,
                                       ln_b, tl_W, tl_b, nodes, Bt1, Bt2, win1,
                                       new_mem, e_proj);
  k_pred<<<500, 256, 0, stream>>>(e_proj, WsT, WdT, ep_bsrc, ep_bdst, ep_Wout,
                                  ep_bout, scores);
  k_mail<<<2 * S4, 128, 0, stream>>>(adj_src, adj_dst, eids, efeat, win2,
                                     new_mem, new_mail);
}